// HNN_80255758893811
// MI455X (gfx1250) — compile-verified
//
#include <hip/hip_runtime.h>

// HNN RK4 rollout for gfx1250 (MI455X).
// Sequential latency-bound problem: persistent per-tile rollout, fp32 WMMA
// (V_WMMA_F32_16X16X4_F32), loop-invariant weight fragments held in VGPRs,
// triple-buffered LDS activation scratch (3 barriers per dynamics eval),
// branch-free tanh on the trans pipe.

#define HID 64

typedef __attribute__((ext_vector_type(2))) float v2f;
typedef __attribute__((ext_vector_type(8))) float v8f;

#define W_STRIDE 65   // padded LDS stride for 64x64 weight staging
#define S_STRIDE 66   // even stride: A-frag pair loads merge to ds_load_b64

static __device__ __forceinline__ v8f wmma4(v2f a, v2f b, v8f c) {
    // D = A(16x4,f32) * B(4x16,f32) + C(16x16,f32)
    return __builtin_amdgcn_wmma_f32_16x16x4_f32(
        /*neg_a=*/false, a, /*neg_b=*/false, b,
        /*c_mod=*/(short)0, c, /*reuse_a=*/false, /*reuse_b=*/false);
}

static __device__ __forceinline__ float fast_tanh(float x) {
    // tanh(x) = 1 - 2/(exp(2x)+1): branch-free, saturates to +/-1 at +/-inf.
    float e = __expf(2.0f * x);                       // v_mul + v_exp_f32
    return 1.0f - 2.0f * __builtin_amdgcn_rcpf(e + 1.0f);
}

__launch_bounds__(128, 1)
__global__ void HNN_rk4_kernel(const float* __restrict__ t,
                               const float* __restrict__ x0,
                               const float* __restrict__ W1,
                               const float* __restrict__ b1,
                               const float* __restrict__ W2,
                               const float* __restrict__ b2,
                               const float* __restrict__ W3,
                               float* __restrict__ out,
                               int Tn, int Bn)
{
    // weight staging (read once into VGPR fragments, then cold)
    __shared__ float sW2 [HID * W_STRIDE];  // sW2 [k*W+n] = W2[k][n]
    __shared__ float sW2T[HID * W_STRIDE];  // sW2T[k*W+n] = W2[n][k]
    __shared__ float sW1T[4 * HID];         // rows 0,1 = W1^T, rows 2,3 = 0 (K pad)
    __shared__ float sW1 [HID * 2];
    __shared__ float sB1 [HID];
    __shared__ float sB2 [HID];
    __shared__ float sW3 [HID];
    // triple-buffered 16x64 activation scratch (D-layout rows)
    __shared__ float sS0 [16 * S_STRIDE];
    __shared__ float sS1 [16 * S_STRIDE];
    __shared__ float sS2 [16 * S_STRIDE];
    __shared__ float sF  [4 * 32];          // per-wave dHdx exchange [16][2]

    const int tid  = threadIdx.x;
    const int wid  = tid >> 5;        // wave 0..3 owns N-tile wid
    const int lane = tid & 31;
    const int nn   = lane & 15;       // n for B/C/D tiles, m for A rows
    const int hi   = lane >> 4;
    const int koff = hi * 2;          // K offset within a K=4 chunk
    const int col  = wid * 16 + nn;

    // ---- stage weights to LDS (coalesced) ----
    for (int i = tid; i < HID * HID; i += 128) {
        int r = i >> 6, c = i & 63;
        float w = W2[i];
        sW2 [r * W_STRIDE + c] = w;
        sW2T[c * W_STRIDE + r] = w;
    }
    for (int i = tid; i < 4 * HID; i += 128) {
        int k = i >> 6, n = i & 63;
        sW1T[i] = (k < 2) ? W1[n * 2 + k] : 0.0f;
    }
    for (int i = tid; i < HID * 2; i += 128) sW1[i] = W1[i];
    for (int i = tid; i < HID; i += 128) { sB1[i] = b1[i]; sB2[i] = b2[i]; sW3[i] = W3[i]; }
    __syncthreads();

    const float b1v = sB1[col];
    const float b2v = sB2[col];
    const float w3v = sW3[col];

    // ---- preload loop-invariant B-matrix fragments into VGPRs ----
    v2f bW1;                                   // stage 1: x @ W1^T (K padded 2->4)
    bW1.x = sW1T[(koff + 0) * HID + col];
    bW1.y = sW1T[(koff + 1) * HID + col];
    v2f bT[16], bN[16], bE[16];
    #pragma unroll
    for (int kk = 0; kk < 16; ++kk) {
        int k0 = kk * 4 + koff;
        bT[kk].x = sW2T[k0 * W_STRIDE + col];        // h1 @ W2^T
        bT[kk].y = sW2T[(k0 + 1) * W_STRIDE + col];
        bN[kk].x = sW2 [k0 * W_STRIDE + col];        // g2 @ W2
        bN[kk].y = sW2 [(k0 + 1) * W_STRIDE + col];
        bE[kk].x = (nn < 2) ? sW1[k0 * 2 + nn] : 0.0f;        // g1 @ W1 (N pad 2->16)
        bE[kk].y = (nn < 2) ? sW1[(k0 + 1) * 2 + nn] : 0.0f;
    }
    float* fWav = &sF[wid * 32];               // this wave's private exchange strip

    // ---- state: 16 batch rows, replicated across half-waves & waves ----
    const int   brow = blockIdx.x * 16 + nn;
    float q = x0[brow * 2 + 0];
    float p = x0[brow * 2 + 1];
    const float dtv = t[1] - t[0];

    auto dynamics = [&](float qq, float pp, float& fq, float& fp) {
        // ---- stage 1: h1 = tanh(x @ W1^T + b1) ----
        v2f ax; ax.x = hi ? 0.0f : qq; ax.y = hi ? 0.0f : pp;
        v8f acc = {};
        acc = wmma4(ax, bW1, acc);
        float h1t[8];
        #pragma unroll
        for (int r = 0; r < 8; ++r) h1t[r] = fast_tanh(acc[r] + b1v);
        #pragma unroll
        for (int r = 0; r < 8; ++r) sS0[(r + 8 * hi) * S_STRIDE + col] = h1t[r];
        __syncthreads();   // B1: h1 visible

        // ---- stage 2: h2 = tanh(h1 @ W2^T + b2); g2 = (1-h2^2)*W3 ----
        v8f aA = {}, aB = {};
        #pragma unroll
        for (int kk = 0; kk < 8; ++kk) {
            int k0 = kk * 4 + koff;
            v2f a;  a.x  = sS0[nn * S_STRIDE + k0];
                    a.y  = sS0[nn * S_STRIDE + k0 + 1];
            aA = wmma4(a, bT[kk], aA);
            int k1 = (kk + 8) * 4 + koff;
            v2f a2; a2.x = sS0[nn * S_STRIDE + k1];
                    a2.y = sS0[nn * S_STRIDE + k1 + 1];
            aB = wmma4(a2, bT[kk + 8], aB);
        }
        #pragma unroll
        for (int r = 0; r < 8; ++r) {
            float h2 = fast_tanh(aA[r] + aB[r] + b2v);
            sS1[(r + 8 * hi) * S_STRIDE + col] = (1.0f - h2 * h2) * w3v;
        }
        __syncthreads();   // B2: g2 visible

        // ---- stage 3: u = g2 @ W2; g1 = (1-h1^2)*u ----
        aA = (v8f){}; aB = (v8f){};
        #pragma unroll
        for (int kk = 0; kk < 8; ++kk) {
            int k0 = kk * 4 + koff;
            v2f a;  a.x  = sS1[nn * S_STRIDE + k0];
                    a.y  = sS1[nn * S_STRIDE + k0 + 1];
            aA = wmma4(a, bN[kk], aA);
            int k1 = (kk + 8) * 4 + koff;
            v2f a2; a2.x = sS1[nn * S_STRIDE + k1];
                    a2.y = sS1[nn * S_STRIDE + k1 + 1];
            aB = wmma4(a2, bN[kk + 8], aB);
        }
        #pragma unroll
        for (int r = 0; r < 8; ++r)
            sS2[(r + 8 * hi) * S_STRIDE + col] =
                (1.0f - h1t[r] * h1t[r]) * (aA[r] + aB[r]);
        __syncthreads();   // B3: g1 visible

        // ---- stage 4: dHdx = g1 @ W1 (redundant on all waves; no extra barrier) ----
        v8f eA = {}, eB = {};
        #pragma unroll
        for (int kk = 0; kk < 8; ++kk) {
            int k0 = kk * 4 + koff;
            v2f a;  a.x  = sS2[nn * S_STRIDE + k0];
                    a.y  = sS2[nn * S_STRIDE + k0 + 1];
            eA = wmma4(a, bE[kk], eA);
            int k1 = (kk + 8) * 4 + koff;
            v2f a2; a2.x = sS2[nn * S_STRIDE + k1];
                    a2.y = sS2[nn * S_STRIDE + k1 + 1];
            eB = wmma4(a2, bE[kk + 8], eB);
        }
        #pragma unroll
        for (int r = 0; r < 8; ++r) {
            float e = eA[r] + eB[r];
            if (nn < 2) fWav[(r + 8 * hi) * 2 + nn] = e;   // dHdx[m][nn]
        }
        __builtin_amdgcn_wave_barrier();   // same-wave LDS ordering fence (free)
        fq =  fWav[nn * 2 + 1];   // dq/dt =  dH/dp
        fp = -fWav[nn * 2 + 0];   // dp/dt = -dH/dq
    };

    // ---- t = 0 ----
    if (tid < 16) {
        out[(size_t)brow * 2 + 0] = q;
        out[(size_t)brow * 2 + 1] = p;
    }

    // ---- RK4 rollout ----
    for (int step = 1; step < Tn; ++step) {
        float k1q, k1p, k2q, k2p, k3q, k3p, k4q, k4p;
        dynamics(q, p, k1q, k1p);
        dynamics(q + 0.5f * dtv * k1q, p + 0.5f * dtv * k1p, k2q, k2p);
        dynamics(q + 0.5f * dtv * k2q, p + 0.5f * dtv * k2p, k3q, k3p);
        dynamics(q + dtv * k3q, p + dtv * k3p, k4q, k4p);
        q += (dtv / 6.0f) * (k1q + 2.0f * k2q + 2.0f * k3q + k4q);
        p += (dtv / 6.0f) * (k1p + 2.0f * k2p + 2.0f * k3p + k4p);
        if (tid < 16) {
            size_t o = ((size_t)step * Bn + brow) * 2;
            out[o + 0] = q;
            out[o + 1] = p;
        }
    }
}

extern "C" void kernel_launch(void* const* d_in, const int* in_sizes, int n_in,
                              void* d_out, int out_size, void* d_ws, size_t ws_size,
                              hipStream_t stream) {
    const float* t  = (const float*)d_in[0];
    const float* x0 = (const float*)d_in[1];
    const float* W1 = (const float*)d_in[2];
    const float* b1 = (const float*)d_in[3];
    const float* W2 = (const float*)d_in[4];
    const float* b2 = (const float*)d_in[5];
    const float* W3 = (const float*)d_in[6];
    // d_in[7] = b3: unused (grad of H independent of b3)
    float* out = (float*)d_out;

    const int Tn = in_sizes[0];          // 256
    const int Bn = in_sizes[1] / 2;      // 2048

    dim3 grid(Bn / 16);
    dim3 block(128);
    hipLaunchKernelGGL(HNN_rk4_kernel, grid, block, 0, stream,
                       t, x0, W1, b1, W2, b2, W3, out, Tn, Bn);
    (void)in_sizes; (void)n_in; (void)out_size; (void)d_ws; (void)ws_size;
}